// FraudDetectionModel_65481071410195
// MI455X (gfx1250) — compile-verified
//
#include <hip/hip_runtime.h>

// ---------------------------------------------------------------------------
// FraudDetectionModel on MI455X (gfx1250, wave32).
// 48 x [Linear(2,2) -> BatchNorm(batch stats) -> tanh*scale+shift] -> Linear(2,1) -> sigmoid
// Structure: per layer, one streaming kernel over the 32MB activation buffer
// (fits in 192MB L2) fused with the previous layer's normalization, plus a
// single-wave stats-reduction kernel (uses V_WMMA_F32_16X16X4_F32 for the
// exact f32 cross-half-wave combine). tanh uses gfx1250's v_tanh_f32.
// ---------------------------------------------------------------------------

typedef __attribute__((ext_vector_type(2))) float v2f;
typedef __attribute__((ext_vector_type(8))) float v8f;

__device__ __forceinline__ float gpu_tanh(float x) {
#if __has_builtin(__builtin_amdgcn_tanhf)
  return __builtin_amdgcn_tanhf(x);
#elif __has_builtin(__builtin_amdgcn_tanh_f32)
  return __builtin_amdgcn_tanh_f32(x);
#else
  float r;
  // v_tanh_f32 is a TRANS op: 1 independent op required before result use.
  asm volatile("v_tanh_f32 %0, %1\n\tv_nop\n\tv_nop" : "=v"(r) : "v"(x));
  return r;
#endif
}

__device__ __forceinline__ float wave_sum32(float v) {
#pragma unroll
  for (int m = 1; m < 32; m <<= 1) v += __shfl_xor(v, m, 32);
  return v;
}

// ---------------------------------------------------------------------------
// Streaming layer kernel. If has_prev: h = tanh(y_prev*A + B)*s + t, else h = x.
// Then y = h @ W^T + b, stored (in-place ok: each thread reads src[j] before
// writing dst[j]; manual 4x batching keeps loads ahead of stores without
// needing (formally illegal) restrict on aliasing pointers).
// Emits block partial sums (Sy0, Sy1, Sy0^2, Sy1^2) -> partials[block*4..].
// ---------------------------------------------------------------------------
__global__ __launch_bounds__(256) void layer_kernel(
    const float2* src, float2* dst,
    const float* __restrict__ Wl, const float* __restrict__ bl,
    const float* __restrict__ cprev, int has_prev,
    float* __restrict__ partials, int n) {
  const float w00 = Wl[0], w01 = Wl[1], w10 = Wl[2], w11 = Wl[3];
  const float b0 = bl[0], b1 = bl[1];
  float A0 = 0.f, A1 = 0.f, B0 = 0.f, B1 = 0.f;
  float s0 = 0.f, s1 = 0.f, t0 = 0.f, t1 = 0.f;
  if (has_prev) {
    A0 = cprev[0]; A1 = cprev[1]; B0 = cprev[2]; B1 = cprev[3];
    s0 = cprev[4]; s1 = cprev[5]; t0 = cprev[6]; t1 = cprev[7];
  }

  float sum0 = 0.f, sum1 = 0.f, sq0 = 0.f, sq1 = 0.f;
  const int stride = gridDim.x * blockDim.x;
  int j = blockIdx.x * blockDim.x + threadIdx.x;

  // main loop: 4 rows in flight per thread (loads batched before stores)
  for (; j + 3 * stride < n; j += 4 * stride) {
    float2 h[4];
#pragma unroll
    for (int u = 0; u < 4; ++u) h[u] = src[j + u * stride];
#pragma unroll
    for (int u = 0; u < 4; ++u) {
      float h0 = h[u].x, h1 = h[u].y;
      if (has_prev) {
        h0 = fmaf(gpu_tanh(fmaf(h0, A0, B0)), s0, t0);
        h1 = fmaf(gpu_tanh(fmaf(h1, A1, B1)), s1, t1);
      }
      float y0 = fmaf(h0, w00, fmaf(h1, w01, b0));
      float y1 = fmaf(h0, w10, fmaf(h1, w11, b1));
      dst[j + u * stride] = make_float2(y0, y1);
      sum0 += y0; sum1 += y1;
      sq0 = fmaf(y0, y0, sq0); sq1 = fmaf(y1, y1, sq1);
    }
  }
  for (; j < n; j += stride) {  // tail
    float2 hv = src[j];
    float h0 = hv.x, h1 = hv.y;
    if (has_prev) {
      h0 = fmaf(gpu_tanh(fmaf(h0, A0, B0)), s0, t0);
      h1 = fmaf(gpu_tanh(fmaf(h1, A1, B1)), s1, t1);
    }
    float y0 = fmaf(h0, w00, fmaf(h1, w01, b0));
    float y1 = fmaf(h0, w10, fmaf(h1, w11, b1));
    dst[j] = make_float2(y0, y1);
    sum0 += y0; sum1 += y1;
    sq0 = fmaf(y0, y0, sq0); sq1 = fmaf(y1, y1, sq1);
  }

  // deterministic block reduction: wave32 shuffle tree, then fixed-order LDS sum
  sum0 = wave_sum32(sum0); sum1 = wave_sum32(sum1);
  sq0  = wave_sum32(sq0);  sq1  = wave_sum32(sq1);
  __shared__ float lds[8][4];
  const int wid = threadIdx.x >> 5, lane = threadIdx.x & 31;
  if (lane == 0) { lds[wid][0] = sum0; lds[wid][1] = sum1; lds[wid][2] = sq0; lds[wid][3] = sq1; }
  __syncthreads();
  if (threadIdx.x == 0) {
    float a = 0.f, b = 0.f, c = 0.f, d = 0.f;
#pragma unroll
    for (int w = 0; w < 8; ++w) { a += lds[w][0]; b += lds[w][1]; c += lds[w][2]; d += lds[w][3]; }
    partials[blockIdx.x * 4 + 0] = a;
    partials[blockIdx.x * 4 + 1] = b;
    partials[blockIdx.x * 4 + 2] = c;
    partials[blockIdx.x * 4 + 3] = d;
  }
}

// ---------------------------------------------------------------------------
// Single-wave stats kernel: total the G block-partials, derive BN constants.
// Cross-half-wave combine done with two chained V_WMMA_F32_16X16X4_F32:
//   A (16x4) selector rows: r0=(1,0,1,0) r1=(0,1,0,1) [wmma1], r2/r3 [wmma2]
//   B (4x16): lanes 0-15 supply K=0,1 (half-0 sums), lanes 16-31 K=2,3 (half-1)
//   => D[c][*] = H0[c] + H1[c] = grand total, exact f32.
// EXEC is all-ones at the WMMA (32 threads, uniform control flow).
// ---------------------------------------------------------------------------
__global__ __launch_bounds__(32) void stats_kernel(
    const float* __restrict__ partials, int G, float inv_n, float eps,
    const float* __restrict__ gamma, const float* __restrict__ beta,
    const float* __restrict__ scale, const float* __restrict__ shift,
    float* __restrict__ cout) {
  const int lane = threadIdx.x;
  float a0 = 0.f, a1 = 0.f, a2 = 0.f, a3 = 0.f;
  for (int j = lane; j < G; j += 32) {
    a0 += partials[4 * j + 0];
    a1 += partials[4 * j + 1];
    a2 += partials[4 * j + 2];
    a3 += partials[4 * j + 3];
  }
  // fold within each 16-lane half (masks 1,2,4,8)
#pragma unroll
  for (int m = 1; m <= 8; m <<= 1) {
    a0 += __shfl_xor(a0, m, 32);
    a1 += __shfl_xor(a1, m, 32);
    a2 += __shfl_xor(a2, m, 32);
    a3 += __shfl_xor(a3, m, 32);
  }
  // WMMA cross-half combine
  const int r = lane & 15;
  v2f sel1 = (r == 0) ? (v2f){1.f, 0.f} : (r == 1) ? (v2f){0.f, 1.f} : (v2f){0.f, 0.f};
  v2f sel2 = (r == 2) ? (v2f){1.f, 0.f} : (r == 3) ? (v2f){0.f, 1.f} : (v2f){0.f, 0.f};
  v2f bv1 = {a0, a1};
  v2f bv2 = {a2, a3};
  v8f d = {0.f, 0.f, 0.f, 0.f, 0.f, 0.f, 0.f, 0.f};
  d = __builtin_amdgcn_wmma_f32_16x16x4_f32(false, sel1, false, bv1, (short)0, d, false, false);
  d = __builtin_amdgcn_wmma_f32_16x16x4_f32(false, sel2, false, bv2, (short)0, d, false, false);

  if (lane == 0) {
    const float S0 = d[0], S1 = d[1], Q0 = d[2], Q1 = d[3];
    const float m0 = S0 * inv_n, m1 = S1 * inv_n;
    const float v0 = fmaf(-m0, m0, Q0 * inv_n);
    const float v1 = fmaf(-m1, m1, Q1 * inv_n);
    const float A0 = gamma[0] * rsqrtf(v0 + eps);
    const float A1 = gamma[1] * rsqrtf(v1 + eps);
    cout[0] = A0;
    cout[1] = A1;
    cout[2] = fmaf(-m0, A0, beta[0]);
    cout[3] = fmaf(-m1, A1, beta[1]);
    cout[4] = scale[0];
    cout[5] = scale[1];
    cout[6] = shift[0];
    cout[7] = shift[1];
  }
}

// ---------------------------------------------------------------------------
// Finalize: apply layer-47 norm+tanh+affine, Linear(2,1), sigmoid.
// sigmoid(z) = 0.5*tanh(z/2)+0.5 (reuses v_tanh_f32).
// ---------------------------------------------------------------------------
__global__ __launch_bounds__(256) void final_kernel(
    const float2* __restrict__ src, const float* __restrict__ cprev,
    const float* __restrict__ Wf, const float* __restrict__ bf,
    float* __restrict__ out, int n) {
  const float A0 = cprev[0], A1 = cprev[1], B0 = cprev[2], B1 = cprev[3];
  const float s0 = cprev[4], s1 = cprev[5], t0 = cprev[6], t1 = cprev[7];
  const float wf0 = Wf[0], wf1 = Wf[1], bfv = bf[0];
  const int stride = gridDim.x * blockDim.x;
  for (int j = blockIdx.x * blockDim.x + threadIdx.x; j < n; j += stride) {
    float2 y = src[j];
    float h0 = fmaf(gpu_tanh(fmaf(y.x, A0, B0)), s0, t0);
    float h1 = fmaf(gpu_tanh(fmaf(y.y, A1, B1)), s1, t1);
    float z = fmaf(h0, wf0, fmaf(h1, wf1, bfv));
    out[j] = fmaf(0.5f, gpu_tanh(0.5f * z), 0.5f);
  }
}

extern "C" void kernel_launch(void* const* d_in, const int* in_sizes, int n_in,
                              void* d_out, int out_size, void* d_ws, size_t ws_size,
                              hipStream_t stream) {
  (void)n_in; (void)out_size; (void)ws_size;
  const float* x     = (const float*)d_in[0];
  const float* W     = (const float*)d_in[1];
  const float* b     = (const float*)d_in[2];
  const float* gamma = (const float*)d_in[3];
  const float* beta  = (const float*)d_in[4];
  const float* scale = (const float*)d_in[5];
  const float* shift = (const float*)d_in[6];
  const float* Wf    = (const float*)d_in[7];
  const float* bf    = (const float*)d_in[8];

  const int n = in_sizes[0] / 2;   // 4,000,000 rows
  const int L = in_sizes[1] / 4;   // 48 layers

  const int G = 1024, BLK = 256;   // 262,144 threads; ~15 rows/thread
  const float inv_n = 1.0f / (float)n;
  const float eps = 1e-5f;

  // workspace layout: [ y buffer n*2 f32 | partials G*4 f32 | constants L*8 f32 ]
  float* ybuf     = (float*)d_ws;
  float* partials = ybuf + (size_t)n * 2;
  float* consts   = partials + (size_t)G * 4;

  // layer 0: read x, write y0, emit partials
  layer_kernel<<<G, BLK, 0, stream>>>((const float2*)x, (float2*)ybuf,
                                      W, b, consts, 0, partials, n);
  stats_kernel<<<1, 32, 0, stream>>>(partials, G, inv_n, eps,
                                     gamma, beta, scale, shift, consts);
  // layers 1..L-1: fused (prev norm+tanh+affine) + linear, in-place on ybuf
  for (int i = 1; i < L; ++i) {
    layer_kernel<<<G, BLK, 0, stream>>>((const float2*)ybuf, (float2*)ybuf,
                                        W + 4 * i, b + 2 * i,
                                        consts + 8 * (i - 1), 1, partials, n);
    stats_kernel<<<1, 32, 0, stream>>>(partials, G, inv_n, eps,
                                       gamma + 2 * i, beta + 2 * i,
                                       scale + 2 * i, shift + 2 * i,
                                       consts + 8 * i);
  }
  // finalize: last norm+tanh+affine, Linear(2,1), sigmoid
  final_kernel<<<G, BLK, 0, stream>>>((const float2*)ybuf, consts + 8 * (L - 1),
                                      Wf, bf, (float*)d_out, n);
}